// BackProjLayer_77395310674506
// MI455X (gfx1250) — compile-verified
//
#include <hip/hip_runtime.h>

#define NCH   32
#define NPX   370
#define S32   33      // padded LDS row stride for 32x32 matrices
#define KPAD  384     // NPX padded to multiple of 16
#define NITER 24      // Newton-Schulz sign iterations

typedef __attribute__((ext_vector_type(2))) float v2f;
typedef __attribute__((ext_vector_type(8))) float v8f;

__device__ __forceinline__ v8f wmma4(v2f a, v2f b, v8f c) {
  // D = A(16x4) * B(4x16) + C, f32 everywhere
  return __builtin_amdgcn_wmma_f32_16x16x4_f32(false, a, false, b, (short)0, c,
                                               false, false);
}

// One wave computes C_tile(m0,n0) += A * B for complex 32x32 matrices in LDS.
__device__ __forceinline__ void cgemm_tile_lds(const float* Ar, const float* Ai,
                                               const float* Br, const float* Bi,
                                               int m0, int n0, v8f& cr, v8f& ci) {
  const int lane = threadIdx.x & 31;
  const int row  = lane & 15;          // M (for A) / N (for B) within tile
  const int kh   = (lane >> 4) << 1;   // K half-select per ISA A/B layout
#pragma unroll
  for (int k0 = 0; k0 < NCH; k0 += 4) {
    v2f ar, ai, br, bi;
#pragma unroll
    for (int v = 0; v < 2; ++v) {
      const int k = k0 + kh + v;
      ar[v] = Ar[(m0 + row) * S32 + k];
      ai[v] = Ai[(m0 + row) * S32 + k];
      br[v] = Br[k * S32 + (n0 + row)];
      bi[v] = Bi[k * S32 + (n0 + row)];
    }
    cr = wmma4(ar, br, cr);    // Re += ArBr
    cr = wmma4(-ai, bi, cr);   // Re -= AiBi
    ci = wmma4(ar, bi, ci);    // Im += ArBi
    ci = wmma4(ai, br, ci);    // Im += AiBr
  }
}

__device__ __forceinline__ void store_tile_lds(float* Cr, float* Ci,
                                               int m0, int n0, v8f cr, v8f ci) {
  const int lane = threadIdx.x & 31;
  const int col  = n0 + (lane & 15);
  const int mb   = m0 + ((lane >> 4) << 3);   // C/D: M = v + 8*(lane/16)
#pragma unroll
  for (int v = 0; v < 8; ++v) {
    Cr[(mb + v) * S32 + col] = cr[v];
    Ci[(mb + v) * S32 + col] = ci[v];
  }
}

__global__ __launch_bounds__(128) void backproj_kernel(
    const float* __restrict__ Sre, const float* __restrict__ Sim,
    const float* __restrict__ tau,
    const float* __restrict__ Dre, const float* __restrict__ Dim,
    const float* __restrict__ Are, const float* __restrict__ Aim,
    float* __restrict__ out,   // (B,32,32) complex64 interleaved
    float* __restrict__ lat)   // (B,370) f32
{
  __shared__ float Hre[NCH * S32], Him[NCH * S32];
  __shared__ float Xre[NCH * S32], Xim[NCH * S32];
  __shared__ float Yre[NCH * S32], Yim[NCH * S32];
  __shared__ float Pre[NCH * S32], Pim[NCH * S32];
  __shared__ float lxs[KPAD];
  __shared__ float red[128];

  const int b    = blockIdx.x;
  const int tid  = threadIdx.x;
  const int wid  = tid >> 5;
  const int lane = tid & 31;
  const int m0   = (wid >> 1) << 4;    // this wave's output tile
  const int n0   = (wid & 1) << 4;
  const int row  = lane & 15;
  const int kh   = (lane >> 4) << 1;
  const size_t base = (size_t)b * (NCH * NCH);

  // ---- Phase 0: Hermitian part H = 0.5*(M + M^H), zero lx pad ----
  for (int e = tid; e < NCH * NCH; e += 128) {
    const int i = e >> 5, j = e & 31;
    const float a = Sre[base + i * 32 + j], at = Sre[base + j * 32 + i];
    const float c = Sim[base + i * 32 + j], ct = Sim[base + j * 32 + i];
    Hre[i * S32 + j] = 0.5f * (a + at);
    Him[i * S32 + j] = 0.5f * (c - ct);
  }
  for (int e = tid; e < KPAD; e += 128) lxs[e] = 0.0f;
  __syncthreads();

  // ---- Frobenius norm -> X0 = H / ||H||_F ----
  float s = 0.0f;
  for (int e = tid; e < NCH * NCH; e += 128) {
    const int i = e >> 5, j = e & 31;
    const float a = Hre[i * S32 + j], c = Him[i * S32 + j];
    s += a * a + c * c;
  }
  red[tid] = s;
  __syncthreads();
  for (int off = 64; off > 0; off >>= 1) {
    if (tid < off) red[tid] += red[tid + off];
    __syncthreads();
  }
  const float inv = rsqrtf(red[0] + 1e-30f);
  for (int e = tid; e < NCH * NCH; e += 128) {
    const int i = e >> 5, j = e & 31;
    Xre[i * S32 + j] = Hre[i * S32 + j] * inv;
    Xim[i * S32 + j] = Him[i * S32 + j] * inv;
  }
  __syncthreads();

  // ---- Phase 1: Newton-Schulz sign iteration  X <- 1.5X - 0.5*X*X*X ----
  for (int it = 0; it < NITER; ++it) {
    v8f yr = {}, yi = {};
    cgemm_tile_lds(Xre, Xim, Xre, Xim, m0, n0, yr, yi);   // Y = X*X
    store_tile_lds(Yre, Yim, m0, n0, yr, yi);
    __syncthreads();
    v8f zr = {}, zi = {};
    cgemm_tile_lds(Xre, Xim, Yre, Yim, m0, n0, zr, zi);   // Z = X*Y
    __syncthreads();
    {
      const int col = n0 + row;
      const int mb  = m0 + ((lane >> 4) << 3);
#pragma unroll
      for (int v = 0; v < 8; ++v) {
        const int idx = (mb + v) * S32 + col;
        Xre[idx] = 1.5f * Xre[idx] - 0.5f * zr[v];
        Xim[idx] = 1.5f * Xim[idx] - 0.5f * zi[v];
      }
    }
    __syncthreads();
  }

  // ---- P = 0.5*(H + H*sign(H))  (PSD projection) ----
  {
    v8f wr = {}, wi = {};
    cgemm_tile_lds(Hre, Him, Xre, Xim, m0, n0, wr, wi);
    const int col = n0 + row;
    const int mb  = m0 + ((lane >> 4) << 3);
#pragma unroll
    for (int v = 0; v < 8; ++v) {
      const int idx = (mb + v) * S32 + col;
      Pre[idx] = 0.5f * (Hre[idx] + wr[v]);
      Pim[idx] = 0.5f * (Him[idx] + wi[v]);
    }
  }
  __syncthreads();

  // ---- Phase 2: latent_x[p] = Re(d_p^H P d_p) - tau[p] via T = P*D ----
  for (int nj = wid; nj < KPAD / 16; nj += 4) {
    const int ncol = nj * 16 + row;          // global pixel column
    const bool okc = (ncol < NPX);
    float colsum = 0.0f;
#pragma unroll
    for (int mi = 0; mi < 2; ++mi) {
      const int pm0 = mi << 4;
      v8f tr = {}, ti = {};
#pragma unroll
      for (int k0 = 0; k0 < NCH; k0 += 4) {
        v2f ar, ai, br, bi;
#pragma unroll
        for (int v = 0; v < 2; ++v) {
          const int k = k0 + kh + v;
          ar[v] = Pre[(pm0 + row) * S32 + k];
          ai[v] = Pim[(pm0 + row) * S32 + k];
          br[v] = okc ? Dre[k * NPX + ncol] : 0.0f;
          bi[v] = okc ? Dim[k * NPX + ncol] : 0.0f;
        }
        tr = wmma4(ar, br, tr);
        tr = wmma4(-ai, bi, tr);
        ti = wmma4(ar, bi, ti);
        ti = wmma4(ai, br, ti);
      }
      if (okc) {
        const int mb = pm0 + ((lane >> 4) << 3);
#pragma unroll
        for (int v = 0; v < 8; ++v)
          colsum += Dre[(mb + v) * NPX + ncol] * tr[v]
                  + Dim[(mb + v) * NPX + ncol] * ti[v];
      }
    }
    colsum += __shfl_xor(colsum, 16, 32);    // combine the two M half-groups
    if (lane < 16 && okc) {
      const float val = colsum - tau[ncol];
      lxs[ncol] = val;
      lat[(size_t)b * NPX + ncol] = val;
    }
  }
  __syncthreads();

  // ---- Phase 3: out = A * diag(lx) * A^H  (K = 370 padded to 384) ----
  v8f orr = {}, oii = {};
  for (int k0 = 0; k0 < KPAD; k0 += 4) {
    v2f asr, asi, br, bi;
#pragma unroll
    for (int v = 0; v < 2; ++v) {
      const int k = k0 + kh + v;
      const bool ok = (k < NPX);
      const float l = lxs[k];               // zero in the pad region
      const float arv = ok ? Are[(m0 + row) * NPX + k] : 0.0f;
      const float aiv = ok ? Aim[(m0 + row) * NPX + k] : 0.0f;
      asr[v] = l * arv;
      asi[v] = l * aiv;
      br[v]  = ok ? Are[(n0 + row) * NPX + k] : 0.0f;
      bi[v]  = ok ? Aim[(n0 + row) * NPX + k] : 0.0f;
    }
    orr = wmma4(asr, br, orr);   // Re += lx*Ar_i*Ar_j
    orr = wmma4(asi, bi, orr);   // Re += lx*Ai_i*Ai_j
    oii = wmma4(asi, br, oii);   // Im += lx*Ai_i*Ar_j
    oii = wmma4(-asr, bi, oii);  // Im -= lx*Ar_i*Ai_j
  }
  {
    float2* o2 = (float2*)out;
    const int col = n0 + row;
    const int mb  = m0 + ((lane >> 4) << 3);
#pragma unroll
    for (int v = 0; v < 8; ++v)
      o2[base + (size_t)(mb + v) * NCH + col] = make_float2(orr[v], oii[v]);
  }
}

extern "C" void kernel_launch(void* const* d_in, const int* in_sizes, int n_in,
                              void* d_out, int out_size, void* d_ws, size_t ws_size,
                              hipStream_t stream) {
  (void)n_in; (void)out_size; (void)d_ws; (void)ws_size;
  const float* Sre = (const float*)d_in[0];
  const float* Sim = (const float*)d_in[1];
  const float* tau = (const float*)d_in[2];
  const float* Dre = (const float*)d_in[3];
  const float* Dim = (const float*)d_in[4];
  const float* Are = (const float*)d_in[5];
  const float* Aim = (const float*)d_in[6];
  const int nB = in_sizes[0] / (NCH * NCH);
  float* out = (float*)d_out;
  float* lat = out + (size_t)nB * NCH * NCH * 2;   // latent_x after complex out
  backproj_kernel<<<nB, 128, 0, stream>>>(Sre, Sim, tau, Dre, Dim, Are, Aim,
                                          out, lat);
}